// BoT_Tokenizer_64269890618106
// MI455X (gfx1250) — compile-verified
//
#include <hip/hip_runtime.h>

typedef __attribute__((ext_vector_type(2))) float v2f;
typedef __attribute__((ext_vector_type(8))) float v8f;

// SINGLE_IDX = [3] + [5..26]
__device__ __constant__ int c_single_idx[23] = {
    3, 5, 6, 7, 8, 9, 10, 11, 12, 13, 14, 15, 16, 17, 18, 19, 20, 21, 22, 23, 24, 25, 26};

// tok_fore (t=0) and tok_palm (t=2) via V_WMMA_F32_16X16X4_F32.
// Block = 256 threads = 8 waves; block handles 16 rows; each wave covers 4
// col-tiles of 16 (wave w -> cols [w*64, w*64+64)). Grid = 8192/16 = 512.
// All gathers select the ADDRESS (index) per half-wave, so each fragment
// element is a single unconditional global_load + v_cndmask on the offset
// (no EXEC predication before the WMMAs, which require EXEC == all-1s).
__global__ __launch_bounds__(256) void fore_palm_wmma_kernel(
    const float* __restrict__ x, const float* __restrict__ Wf,
    const float* __restrict__ bf, const float* __restrict__ Wp,
    const float* __restrict__ bp, float* __restrict__ out) {
  const int lane = threadIdx.x & 31;
  const int wave = threadIdx.x >> 5;
  const int half = lane >> 4;  // 0: lanes 0-15, 1: lanes 16-31
  const int l16 = lane & 15;
  const int rowBase = blockIdx.x * 16;
  const float* xr = x + (rowBase + l16) * 39;

  // A fragments (16x4 f32): lanes<16 hold K={k0,k0+1}, lanes>=16 K={k0+2,k0+3}.
  // FORE_IDX = {0,1,2,27,28,32,33,34,38}, K padded 9->12 with zeros.
  v2f af0, af1, af2;
  af0.x = xr[half ? 2 : 0];
  af0.y = xr[half ? 27 : 1];
  af1.x = xr[half ? 33 : 28];
  af1.y = xr[half ? 34 : 32];
  const float x38 = xr[38];            // unconditional load, then value-select
  af2.x = half ? 0.0f : x38;
  af2.y = 0.0f;
  // PALM_IDX = {4,29,30,31,35,36,37}, K padded 7->8.
  v2f ap0, ap1;
  ap0.x = xr[half ? 30 : 4];
  ap0.y = xr[half ? 31 : 29];
  ap1.x = xr[half ? 37 : 35];
  const float x36 = xr[36];
  ap1.y = half ? 0.0f : x36;

#pragma unroll
  for (int j = 0; j < 4; ++j) {
    const int colBase = wave * 64 + j * 16;
    const int n = colBase + l16;

    // B = Wf^T (4x16 tile): VGPR0 = rows K={k0,k0+2}, VGPR1 = K={k0+1,k0+3}
    const float* wfn = Wf + n * 9;
    v2f b0, b1, b2;
    b0.x = wfn[half ? 2 : 0];
    b0.y = wfn[half ? 3 : 1];
    b1.x = wfn[half ? 6 : 4];
    b1.y = wfn[half ? 7 : 5];
    const float w8 = wfn[8];
    b2.x = half ? 0.0f : w8;
    b2.y = 0.0f;

    const float biasf = bf[n];
    v8f c;
#pragma unroll
    for (int v = 0; v < 8; ++v) c[v] = biasf;  // C[m,n] = bf[n]
    c = __builtin_amdgcn_wmma_f32_16x16x4_f32(false, af0, false, b0, (short)0, c, false, false);
    c = __builtin_amdgcn_wmma_f32_16x16x4_f32(false, af1, false, b1, (short)0, c, false, false);
    c = __builtin_amdgcn_wmma_f32_16x16x4_f32(false, af2, false, b2, (short)0, c, false, false);
    // C/D layout: VGPR v -> M = v + 8*half, N = colBase + l16; token slot t=0
#pragma unroll
    for (int v = 0; v < 8; ++v) {
      const int m = v + half * 8;
      out[((size_t)(rowBase + m) * 25 + 0) * 512 + n] = c[v];
    }

    // Palm: K=8 (2 WMMAs), token slot t=2
    const float* wpn = Wp + n * 7;
    v2f p0, p1;
    p0.x = wpn[half ? 2 : 0];
    p0.y = wpn[half ? 3 : 1];
    p1.x = wpn[half ? 6 : 4];
    const float w5 = wpn[5];
    p1.y = half ? 0.0f : w5;

    const float biasp = bp[n];
    v8f cp;
#pragma unroll
    for (int v = 0; v < 8; ++v) cp[v] = biasp;
    cp = __builtin_amdgcn_wmma_f32_16x16x4_f32(false, ap0, false, p0, (short)0, cp, false, false);
    cp = __builtin_amdgcn_wmma_f32_16x16x4_f32(false, ap1, false, p1, (short)0, cp, false, false);
#pragma unroll
    for (int v = 0; v < 8; ++v) {
      const int m = v + half * 8;
      out[((size_t)(rowBase + m) * 25 + 2) * 512 + n] = cp[v];
    }
  }
}

// tok_single: out[row, t(s), c] = x[row, SINGLE_IDX[s]] * Ws[s,c] + bs[s,c]
// s==0 -> t=1, s>=1 -> t=s+2. One thread = 4 contiguous channels (float4 path).
// This kernel covers 92% of the 419 MB output -> pure streaming, coalesced.
__global__ __launch_bounds__(256) void single_token_kernel(
    const float* __restrict__ x, const float* __restrict__ Ws,
    const float* __restrict__ bs, float* __restrict__ out) {
  const unsigned gid = blockIdx.x * 256u + threadIdx.x;
  const int q = gid & 127;        // 512/4 = 128 quads per (row, s)
  const unsigned rs = gid >> 7;   // row*23 + s
  const int s = rs % 23u;
  const int row = rs / 23u;
  const int c0 = q * 4;

  const float xv = x[row * 39 + c_single_idx[s]];
  const float4 w = *(const float4*)(Ws + s * 512 + c0);
  const float4 b = *(const float4*)(bs + s * 512 + c0);
  float4 r;
  r.x = fmaf(xv, w.x, b.x);
  r.y = fmaf(xv, w.y, b.y);
  r.z = fmaf(xv, w.z, b.z);
  r.w = fmaf(xv, w.w, b.w);

  const int t = (s == 0) ? 1 : (s + 2);
  *(float4*)(out + ((size_t)row * 25 + t) * 512 + c0) = r;
}

extern "C" void kernel_launch(void* const* d_in, const int* in_sizes, int n_in,
                              void* d_out, int out_size, void* d_ws, size_t ws_size,
                              hipStream_t stream) {
  const float* x = (const float*)d_in[0];    // (8192, 39)
  const float* Wf = (const float*)d_in[1];   // (512, 9)
  const float* bf = (const float*)d_in[2];   // (512,)
  const float* Wp = (const float*)d_in[3];   // (512, 7)
  const float* bp = (const float*)d_in[4];   // (512,)
  const float* Ws = (const float*)d_in[5];   // (23, 512)
  const float* bs = (const float*)d_in[6];   // (23, 512)
  float* out = (float*)d_out;                // (8192, 25, 512)

  fore_palm_wmma_kernel<<<8192 / 16, 256, 0, stream>>>(x, Wf, bf, Wp, bp, out);

  const unsigned totalQuads = 8192u * 23u * 128u;  // 24,117,248
  single_token_kernel<<<totalQuads / 256u, 256, 0, stream>>>(x, Ws, bs, out);
}